// BatchedEBSDSuper_51926154609196
// MI455X (gfx1250) — compile-verified
//
#include <hip/hip_runtime.h>

// ============================================================================
// BatchedEBSDSuper on MI455X (gfx1250)
//
// Strategy: fold all 8 real Wigner-3j tensors + tp_weights + alpha into a
// single 484x88 matrix T ( (i,j) channel-pair -> copy*22+m output ), then
//   out[n, :] = vec(x1[n] (outer) x2[n]) @ T          (+ residual)
// computed as an fp32 WMMA GEMM (V_WMMA_F32_16X16X4_F32, K=484=121*4 exact).
// fp32 WMMA keeps the K=484 accumulation bit-compatible with the fp32
// reference (f16/bf16 operands would give ~1e-2 rel error).
// x2 is the 3x3 edge-padded neighbor sum, produced by a separate kernel.
//
// Main kernel: each wave owns TWO 16-pixel M-tiles (32 pixels), so every
// B-operand LDS fetch and all k-index math feeds 12 WMMAs. T is staged once
// per block in LDS (186 KB of the 320 KB WGP LDS -> 1 WG/WGP, latency hidden
// by the 12-way independent WMMA chain per k-step).
//
// Workspace layout (floats):
//   [0      , 10648) : 8 real Wigner tensors (generated on device, fp64 math)
//   [10648  , 21296) : imag-part scratch for the basis change
//   [21296  , 67760) : packed WMMA-B matrix T (121 ksteps x 6 ntiles x 2 x 32)
//   [67760  , 67760 + N*22) : neighbor-sum field nb  (~46 MB for N=524288)
// ============================================================================

typedef __attribute__((ext_vector_type(2))) float v2f;
typedef __attribute__((ext_vector_type(8))) float v8f;

#define NTILE    6
#define KSTEPS   121
#define NOUT     88
#define TB_FLOATS (KSTEPS * NTILE * 64)   // 46464

#define WS_WIG    0
#define WS_WIGIM  10648
#define WS_TB     21296
#define WS_NB     67760

#define ALPHA4 1.5f
#define ALPHA6 1.80277563773199464656f    // sqrt(13)/2

// ---------------------------------------------------------------------------
// Device Wigner-3j generation (Racah formula + complex->real basis change)
// ---------------------------------------------------------------------------
__device__ __constant__ double c_fact[21] = {
    1.0, 1.0, 2.0, 6.0, 24.0, 120.0, 720.0, 5040.0, 40320.0,
    362880.0, 3628800.0, 39916800.0, 479001600.0, 6227020800.0,
    87178291200.0, 1307674368000.0, 20922789888000.0, 355687428096000.0,
    6402373705728000.0, 121645100408832000.0, 2432902008176640000.0};

__device__ inline double cg_racah(int l1, int l2, int l3, int m1, int m2) {
    int m3 = m1 + m2;
    if (m3 < -l3 || m3 > l3) return 0.0;
    double pref0 = sqrt((double)(2 * l3 + 1) * c_fact[l1 + l2 - l3] *
                        c_fact[l1 - l2 + l3] * c_fact[-l1 + l2 + l3] /
                        c_fact[l1 + l2 + l3 + 1]);
    double pref = pref0 * sqrt(c_fact[l1 + m1] * c_fact[l1 - m1] *
                               c_fact[l2 + m2] * c_fact[l2 - m2] *
                               c_fact[l3 + m3] * c_fact[l3 - m3]);
    int kmin = max(0, max(l2 - l3 - m1, l1 - l3 + m2));
    int kmax = min(l1 + l2 - l3, min(l1 - m1, l2 + m2));
    double s = 0.0;
    for (int k = kmin; k <= kmax; ++k) {
        double d = c_fact[k] * c_fact[l1 + l2 - l3 - k] * c_fact[l1 - m1 - k] *
                   c_fact[l2 + m2 - k] * c_fact[l3 - l2 + m1 + k] *
                   c_fact[l3 - l1 - m2 + k];
        s += ((k & 1) ? -1.0 : 1.0) / d;
    }
    return pref * s;
}

struct URow { int n; int col[2]; double re[2]; double im[2]; };

// Row 'a' of the complex->real spherical harmonic change-of-basis U(l)
__device__ inline URow urow(int l, int a) {
    URow r;
    const double is2 = 0.70710678118654752440;
    int m = a - l;
    if (m == 0) {
        r.n = 1; r.col[0] = l; r.re[0] = 1.0; r.im[0] = 0.0;
        r.col[1] = l; r.re[1] = 0.0; r.im[1] = 0.0;
    } else if (m > 0) {
        r.n = 2;
        r.col[0] = l + m; r.re[0] = ((m & 1) ? -is2 : is2); r.im[0] = 0.0;
        r.col[1] = l - m; r.re[1] = is2;                    r.im[1] = 0.0;
    } else {
        int mm = -m; r.n = 2;
        r.col[0] = l - mm; r.re[0] = 0.0; r.im[0] = is2;                      // i/sqrt2
        r.col[1] = l + mm; r.re[1] = 0.0; r.im[1] = ((mm & 1) ? is2 : -is2);  // -i(-1)^m/sqrt2
    }
    return r;
}

// One block per (l1,l2,l3) tensor; Frobenius-normalized real intertwiner.
__global__ void wigner_kernel(float* __restrict__ out, float* __restrict__ tmp) {
    const int L1[8] = {4, 4, 6, 6, 4, 4, 6, 6};
    const int L2[8] = {4, 6, 4, 6, 4, 6, 4, 6};
    const int L3[8] = {4, 4, 4, 4, 6, 6, 6, 6};
    const int OFF[8] = {0, 729, 1782, 2835, 4356, 5409, 6930, 8451};
    const int blk = blockIdx.x;
    const int l1 = L1[blk], l2 = L2[blk], l3 = L3[blk];
    const int d2 = 2 * l2 + 1, d3 = 2 * l3 + 1;
    const int sz = (2 * l1 + 1) * d2 * d3;
    float* o  = out + OFF[blk];
    float* oi = tmp + OFF[blk];

    double sr = 0.0, si = 0.0;
    for (int e = threadIdx.x; e < sz; e += blockDim.x) {
        int cc = e % d3;
        int r  = e / d3;
        int bb = r % d2;
        int aa = r / d2;
        URow u1 = urow(l1, aa), u2 = urow(l2, bb), u3 = urow(l3, cc);
        double kr = 0.0, ki = 0.0;
        for (int i1 = 0; i1 < u1.n; ++i1)
            for (int i2 = 0; i2 < u2.n; ++i2)
                for (int i3 = 0; i3 < u3.n; ++i3) {
                    int m1 = u1.col[i1] - l1;
                    int m2 = u2.col[i2] - l2;
                    int m3 = u3.col[i3] - l3;
                    if (m1 + m2 != m3) continue;
                    double C = cg_racah(l1, l2, l3, m1, m2);
                    if (C == 0.0) continue;
                    double tr = u1.re[i1] * u2.re[i2] - u1.im[i1] * u2.im[i2];
                    double ti = u1.re[i1] * u2.im[i2] + u1.im[i1] * u2.re[i2];
                    double wr = tr * u3.re[i3] + ti * u3.im[i3];
                    double wi = ti * u3.re[i3] - tr * u3.im[i3];
                    kr += wr * C;
                    ki += wi * C;
                }
        o[e]  = (float)kr;
        oi[e] = (float)ki;
        sr += kr * kr;
        si += ki * ki;
    }

    __shared__ double shr[256];
    __shared__ double shi[256];
    shr[threadIdx.x] = sr;
    shi[threadIdx.x] = si;
    __syncthreads();
    for (int s = 128; s > 0; s >>= 1) {
        if ((int)threadIdx.x < s) {
            shr[threadIdx.x] += shr[threadIdx.x + s];
            shi[threadIdx.x] += shi[threadIdx.x + s];
        }
        __syncthreads();
    }
    const double nr = sqrt(shr[0]);
    const double ni = sqrt(shi[0]);
    const bool useIm = (nr < 1e-8);
    const double inv = 1.0 / (useIm ? ni : nr);
    for (int e = threadIdx.x; e < sz; e += blockDim.x) {
        double v = useIm ? (double)oi[e] : (double)o[e];
        o[e] = (float)(v * inv);
    }
}

// ---------------------------------------------------------------------------
// Build packed WMMA-B matrix: T[ks][q][v][lane], K = v + (lane<16?0:2),
// n = q*16 + lane%16.  Folds tp_weights and alpha factors.
// ---------------------------------------------------------------------------
__device__ __constant__ int c_off4[4] = {0, 729, 1782, 2835};       // W444 W464 W644 W664
__device__ __constant__ int c_off6[4] = {4356, 5409, 6930, 8451};   // W446 W466 W646 W666
__device__ __constant__ int c_jd[4]   = {9, 13, 9, 13};

__global__ void build_tb_kernel(const float* __restrict__ wig,
                                const float* __restrict__ tp,
                                float* __restrict__ TB) {
    int idx = blockIdx.x * blockDim.x + threadIdx.x;
    if (idx >= TB_FLOATS) return;
    const int lane = idx & 31;
    const int v    = (idx >> 5) & 1;
    const int q    = (idx >> 6) % NTILE;
    const int ks   = idx / (NTILE * 64);
    const int K = v + ((lane < 16) ? 0 : 2);
    const int k = ks * 4 + K;          // (i,j) flat, i = x1 channel, j = x2 channel
    const int n = q * 16 + (lane & 15);
    float val = 0.0f;
    if (k < 484 && n < NOUT) {
        const int i = k / 22, j = k % 22;
        const int cpy = n / 22, m = n % 22;
        const int p  = ((i >= 9) ? 2 : 0) + ((j >= 9) ? 1 : 0);
        const int ii = i - ((i >= 9) ? 9 : 0);
        const int jj = j - ((j >= 9) ? 9 : 0);
        if (m < 9) {
            val = ALPHA4 * tp[cpy * 8 + 2 * p] *
                  wig[c_off4[p] + (ii * c_jd[p] + jj) * 9 + m];
        } else {
            val = ALPHA6 * tp[cpy * 8 + 2 * p + 1] *
                  wig[c_off6[p] + (ii * c_jd[p] + jj) * 13 + (m - 9)];
        }
    }
    TB[idx] = val;
}

// ---------------------------------------------------------------------------
// 3x3 edge-padded neighbor sum -> nb[N][22]
// ---------------------------------------------------------------------------
__global__ void conv_kernel(const float* __restrict__ f4,
                            const float* __restrict__ f6,
                            const float* __restrict__ w,
                            const int* __restrict__ pH,
                            const int* __restrict__ pW,
                            float* __restrict__ nb, int N) {
    const long long tid = (long long)blockIdx.x * blockDim.x + threadIdx.x;
    const long long total = (long long)N * 22;
    if (tid >= total) return;
    const int H = *pH, W = *pW;
    const int c = (int)(tid % 22);
    const long long n = tid / 22;
    const int x = (int)(n % W);
    const long long r = n / W;
    const int y = (int)(r % H);
    const long long b = r / H;
    float acc = 0.0f;
#pragma unroll
    for (int dy = 0; dy < 3; ++dy) {
        int yy = y + dy - 1;
        yy = yy < 0 ? 0 : (yy >= H ? H - 1 : yy);
#pragma unroll
        for (int dx = 0; dx < 3; ++dx) {
            int xx = x + dx - 1;
            xx = xx < 0 ? 0 : (xx >= W ? W - 1 : xx);
            const long long nn = (b * H + yy) * (long long)W + xx;
            const float vv = (c < 9) ? f4[nn * 9 + c] : f6[nn * 13 + (c - 9)];
            acc += w[dy * 3 + dx] * vv;
        }
    }
    nb[n * 22 + c] = acc;
}

// ---------------------------------------------------------------------------
// Main fused WMMA GEMM: each wave = 2 M-tiles (32 pixels) x (484 -> 88).
// ---------------------------------------------------------------------------
__global__ __launch_bounds__(128) void tp_wmma_kernel(
    const float* __restrict__ f4, const float* __restrict__ f6,
    const float* __restrict__ nb, const float* __restrict__ TBg,
    float* __restrict__ out, int N) {
    extern __shared__ float smem[];
    float* sTB = smem;                      // TB_FLOATS
    float* sXY = smem + TB_FLOATS;          // 4 waves * (704 x + 704 y)

    const int tid  = threadIdx.x;
    const int wave = tid >> 5;
    const int lane = tid & 31;

    for (int i = tid; i < TB_FLOATS; i += 128) sTB[i] = TBg[i];

    float* xw = sXY + wave * 1408;
    float* yw = xw + 704;

    const int m    = lane & 15;             // A-matrix row within each tile
    const int koff = (lane < 16) ? 0 : 2;   // A-matrix K group per lane half
    const int nWG  = N >> 7;                // 128 pixels per workgroup-tile

    for (int t = blockIdx.x; t < nWG; t += gridDim.x) {
        const int base = (t << 7) + (wave << 5);   // 32 pixels per wave
        __syncthreads();                    // TB ready / protect LDS reuse
        for (int idx = lane; idx < 704; idx += 32) {
            const int c  = idx >> 5;        // channel 0..21
            const int mm = idx & 31;        // pixel in wave tile
            const int n  = base + mm;
            xw[idx] = (c < 9) ? f4[n * 9 + c] : f6[n * 13 + (c - 9)];
            yw[idx] = nb[n * 22 + c];
        }
        if (t + 1 < nWG)  // warm caches for the next tile's smoothed field
            __builtin_prefetch(&nb[(long long)(base + 128) * 22], 0, 1);
        __syncthreads();

        v8f acc0[NTILE], acc1[NTILE];
        const v8f zero = {0.f, 0.f, 0.f, 0.f, 0.f, 0.f, 0.f, 0.f};
#pragma unroll
        for (int q = 0; q < NTILE; ++q) { acc0[q] = zero; acc1[q] = zero; }

        // Incrementally tracked (i,j): k0 = ks*4 + koff is always even, so
        // j0 is even and (i,j0) == (i,j0+1) -> one x value feeds both K's.
        int ii = 0, jj = koff;
        const float* tb = sTB;
#pragma unroll 1
        for (int ks = 0; ks < KSTEPS; ++ks) {
            const float* xr = xw + ii * 32;
            const float* yr = yw + jj * 32;
            const float x0  = xr[m];
            const float x1  = xr[m + 16];
            const float y00 = yr[m];
            const float y01 = yr[32 + m];
            const float y10 = yr[m + 16];
            const float y11 = yr[32 + m + 16];
            v2f a0, a1;
            a0.x = x0 * y00; a0.y = x0 * y01;
            a1.x = x1 * y10; a1.y = x1 * y11;
#pragma unroll
            for (int q = 0; q < NTILE; ++q) {
                v2f b;
                b.x = tb[q * 64 + lane];
                b.y = tb[q * 64 + 32 + lane];
                acc0[q] = __builtin_amdgcn_wmma_f32_16x16x4_f32(
                    false, a0, false, b, (short)0, acc0[q], false, false);
                acc1[q] = __builtin_amdgcn_wmma_f32_16x16x4_f32(
                    false, a1, false, b, (short)0, acc1[q], false, false);
            }
            tb += NTILE * 64;
            jj += 4;
            if (jj >= 22) { jj -= 22; ++ii; }
        }

        // Epilogue: D layout -> VGPR r: M = r + (lane<16?0:8), Nout = lane%16
        const int col0  = lane & 15;
        const int mhalf = (lane < 16) ? 0 : 8;
        auto epilogue = [&](const v8f* acc, int tbase) {
#pragma unroll
            for (int q = 0; q < NTILE; ++q) {
                const int col = q * 16 + col0;
                if (col < NOUT) {
#pragma unroll
                    for (int r = 0; r < 8; ++r) {
                        const int px = tbase + r + mhalf;
                        const float v = acc[q][r];
                        if (col < 36) {
                            out[(long long)px * 36 + col] = v + f4[px * 9 + (col % 9)];
                        } else {
                            const int c6 = col - 36;
                            out[(long long)N * 36 + (long long)px * 52 + c6] =
                                v + f6[px * 13 + (c6 % 13)];
                        }
                    }
                }
            }
        };
        epilogue(acc0, base);
        epilogue(acc1, base + 16);
    }
}

// ---------------------------------------------------------------------------
extern "C" void kernel_launch(void* const* d_in, const int* in_sizes, int n_in,
                              void* d_out, int out_size, void* d_ws, size_t ws_size,
                              hipStream_t stream) {
    const float* f4 = (const float*)d_in[0];
    const float* f6 = (const float*)d_in[1];
    const float* sw = (const float*)d_in[2];
    const float* tp = (const float*)d_in[3];
    const int* pH = (const int*)d_in[4];
    const int* pW = (const int*)d_in[5];
    // d_in[6] = batch_size (unused: derived implicitly)

    const int N = in_sizes[0] / 9;  // 524288 for the reference setup

    float* ws    = (float*)d_ws;
    float* wig   = ws + WS_WIG;
    float* wigIm = ws + WS_WIGIM;
    float* TB    = ws + WS_TB;
    float* nb    = ws + WS_NB;      // requires ws_size >= ~46.5 MB

    wigner_kernel<<<8, 256, 0, stream>>>(wig, wigIm);
    build_tb_kernel<<<(TB_FLOATS + 255) / 256, 256, 0, stream>>>(wig, tp, TB);

    {
        const long long total = (long long)N * 22;
        const int blocks = (int)((total + 255) / 256);
        conv_kernel<<<blocks, 256, 0, stream>>>(f4, f6, sw, pH, pW, nb, N);
    }
    {
        const int nWG = N >> 7;     // 128 pixels per workgroup
        int blocks = (nWG < 2048) ? (nWG > 0 ? nWG : 1) : 2048;
        const size_t shmem = ((size_t)TB_FLOATS + 4 * 1408) * 4;  // ~208 KB
        tp_wmma_kernel<<<blocks, 128, shmem, stream>>>(f4, f6, nb, TB,
                                                       (float*)d_out, N);
    }
}